// DiscriminativeLoss_46677704573718
// MI455X (gfx1250) — compile-verified
//
#include <hip/hip_runtime.h>
#include <hip/hip_bf16.h>
#include <math.h>

// ---------------------------------------------------------------------------
// DiscriminativeLoss for MI455X (gfx1250, wave32).
//
// B=4, nf=16, H=W=512 -> N=262144 pixels/batch, C=8 instances.
// Memory-bound: 2 passes over pred (2x64MB) + target (2x4MB) ~ 136MB
// -> ~6us at 23.3 TB/s. Pass 1's einsum('bfn,bnc->bfc') is a 16xN * Nx8
// matmul with a one-hot B matrix: done with V_WMMA_F32_16X16X4_F32.
// A second WMMA with an all-ones A-matrix yields the per-class pixel
// counts (every row of ones x mask = counts), avoiding any divergent
// per-lane histogram code in the hot loop.
// ---------------------------------------------------------------------------

typedef __attribute__((ext_vector_type(2))) float v2f;
typedef __attribute__((ext_vector_type(8))) float v8f;

#define NPIX   (512 * 512)
#define NB     4
#define NF     16
#define NC     8
#define DELTA_V 0.5f
#define CLAMP_MAX 100000.0f

// ---- kernel 0: zero workspace accumulators --------------------------------
__global__ void dl_zero_kernel(float* __restrict__ ws, int n) {
    int i = blockIdx.x * blockDim.x + threadIdx.x;
    if (i < n) ws[i] = 0.0f;
}

// ---- kernel 1: per-class feature sums + counts via WMMA f32 16x16x4 -------
// grid = (64, B), block = 256 (8 waves). Each wave owns a contiguous chunk
// of pixels and performs chunk/4 WMMA accumulations per accumulator.
//
// A fragment (16 feats x 4 pix, f32): lane L holds feature M=L&15;
//   lanes 0-15: vgpr0/1 = pixels k0,k1 ; lanes 16-31: vgpr0/1 = pixels k2,k3.
// B fragment (4 pix x 16 "classes"): lane L holds class Ncol=L&15;
//   lanes 0-15: vgpr0/1 = mask(k0),mask(k1); lanes 16-31: mask(k2),mask(k3).
// C/D (16x16 f32): lane L, vgpr r -> row M = r + 8*(L>=16), col N = L&15.
__global__ void dl_sums_kernel(const float* __restrict__ pred,
                               const int*   __restrict__ gt,
                               float* __restrict__ sums,    // [B][16][8]
                               float* __restrict__ counts,  // [B][8]
                               int pix_per_wave) {
    const int b    = blockIdx.y;
    const int lane = threadIdx.x & 31;
    const int wave = blockIdx.x * (blockDim.x >> 5) + (threadIdx.x >> 5);
    const int f    = lane & 15;      // feature row (A) / class col (B)
    const int hi   = lane >> 4;      // 0 = K pair {0,1}, 1 = K pair {2,3}

    const float* predf = pred + ((size_t)b * NF + f) * NPIX;
    const int*   gtb   = gt   + (size_t)b * NPIX;
    const int    base0 = wave * pix_per_wave;

    v8f acc  = {};   // sums[feature][class]
    v8f accn = {};   // counts[class] replicated in every row
    const v2f ones = {1.0f, 1.0f};

    for (int base = base0; base < base0 + pix_per_wave; base += 4) {
        // A: 4 consecutive pixels of this lane's feature row (b128 load).
        const float4 p4 = *(const float4*)(predf + base);
        v2f a;
        a.x = hi ? p4.z : p4.x;
        a.y = hi ? p4.w : p4.y;

        // B: one-hot mask built from two gt labels (no mask tensor in memory).
        const int2 g2 = *(const int2*)(gtb + base + 2 * hi);
        v2f bm;
        bm.x = (g2.x == f) ? 1.0f : 0.0f;
        bm.y = (g2.y == f) ? 1.0f : 0.0f;

        // sums  += pred(16x4) * mask(4x16)
        acc = __builtin_amdgcn_wmma_f32_16x16x4_f32(
            false, a, false, bm, (short)0, acc, false, false);
        // counts += ones(16x4) * mask(4x16)  (each row == per-class count)
        accn = __builtin_amdgcn_wmma_f32_16x16x4_f32(
            false, ones, false, bm, (short)0, accn, false, false);
    }

    // Merge accumulators into global memory (cols 8..15 are identically 0).
    if (f < NC) {
#pragma unroll
        for (int r = 0; r < 8; ++r) {
            const int row = r + 8 * hi;  // feature index
            atomicAdd(&sums[((size_t)b * NF + row) * NC + f], acc[r]);
        }
        if (hi == 0)  // row 0 of the ones-product: lanes 0..7, vgpr 0
            atomicAdd(&counts[b * NC + f], accn[0]);
    }
}

// ---- kernel 2: per-pixel hinge-distance accumulation ----------------------
__global__ void dl_dist_kernel(const float* __restrict__ pred,
                               const int*   __restrict__ gt,
                               const float* __restrict__ sums,
                               const float* __restrict__ counts,
                               float* __restrict__ dist) {  // [B]
    __shared__ float means[NF * NC];
    __shared__ float red[256];
    const int b   = blockIdx.y;
    const int tid = threadIdx.x;

    if (tid < NF * NC)
        means[tid] = sums[b * NF * NC + tid] / counts[b * NC + (tid & (NC - 1))];
    __syncthreads();

    const float* predb = pred + (size_t)b * NF * NPIX;
    const int*   gtb   = gt   + (size_t)b * NPIX;
    const int stride   = gridDim.x * blockDim.x;

    float acc = 0.0f;
    for (int n = blockIdx.x * blockDim.x + tid; n < NPIX; n += stride) {
        const int g = gtb[n];
        __builtin_prefetch(gtb + n + stride, 0, 1);  // global_prefetch_b8
        float sq = 0.0f;
#pragma unroll
        for (int fe = 0; fe < NF; ++fe) {
            const float d = means[fe * NC + g] - predb[(size_t)fe * NPIX + n];
            sq = fmaf(d, d, sq);
        }
        const float dd = sqrtf(sq);  // sqrt(0)=0 matches the guarded sqrt
        const float t  = fminf(fmaxf(dd - DELTA_V, 0.0f), CLAMP_MAX);
        acc = fmaf(t, t, acc);
    }

    red[tid] = acc;
    __syncthreads();
    for (int s = 128; s > 0; s >>= 1) {
        if (tid < s) red[tid] += red[tid + s];
        __syncthreads();
    }
    if (tid == 0) atomicAdd(&dist[b], red[0]);
}

// ---- kernel 3: finalize scalar loss ---------------------------------------
__global__ void dl_final_kernel(const float* __restrict__ counts,
                                const float* __restrict__ dist,
                                float* __restrict__ out) {
    float l = 0.0f;
    const int b = threadIdx.x;
    if (b < NB) {
        float inv = 0.0f;
#pragma unroll
        for (int c = 0; c < NC; ++c) inv += 1.0f / counts[b * NC + c];
        l = dist[b] * inv * (1.0f / (float)NC);
    }
    l += __shfl_down(l, 2);
    l += __shfl_down(l, 1);
    if (threadIdx.x == 0) out[0] = l;  // ALPHA = 1
}

// ---------------------------------------------------------------------------
extern "C" void kernel_launch(void* const* d_in, const int* in_sizes, int n_in,
                              void* d_out, int out_size, void* d_ws, size_t ws_size,
                              hipStream_t stream) {
    const float* pred = (const float*)d_in[0];  // [4,16,512,512] f32
    const int*   gt   = (const int*)d_in[1];    // [4,512,512] i32

    float* ws     = (float*)d_ws;
    float* sums   = ws;                       // B*16*8 = 512 floats
    float* counts = sums + NB * NF * NC;      // B*8    =  32 floats
    float* dist   = counts + NB * NC;         // B      =   4 floats
    const int n_accum = NB * NF * NC + NB * NC + NB;  // 548

    dl_zero_kernel<<<dim3((n_accum + 255) / 256), 256, 0, stream>>>(ws, n_accum);

    // 64 blocks x 8 waves per batch -> 512 waves, 512 pixels each.
    const int waves_per_batch = 64 * (256 / 32);
    dl_sums_kernel<<<dim3(64, NB), 256, 0, stream>>>(
        pred, gt, sums, counts, NPIX / waves_per_batch);

    dl_dist_kernel<<<dim3(128, NB), 256, 0, stream>>>(pred, gt, sums, counts, dist);

    dl_final_kernel<<<1, 32, 0, stream>>>(counts, dist, (float*)d_out);
}